// MultiHeadAttentionBlock_13597866459702
// MI455X (gfx1250) — compile-verified
//
#include <hip/hip_runtime.h>
#include <hip/hip_bf16.h>

#define B_  2
#define S_  2048
#define H_  8
#define DM_ 256
#define DH_ 32
#define NT_ 21   // 2*MAX_REL+1

typedef __attribute__((ext_vector_type(2))) float v2f;
typedef __attribute__((ext_vector_type(8))) float v8f;

// ---------------------------------------------------------------------------
// Y[4096,256] = X[4096,256] @ W^T + bias   (torch Linear), fp32 WMMA 16x16x4.
// Block: 256 threads = 8 waves. Wave w -> rows [blk.x*128 + 16w, +16),
// cols [blk.y*64, +64) as 4 accumulators sharing one A fragment.
// ---------------------------------------------------------------------------
__global__ __launch_bounds__(256) void proj_wmma_kernel(
    const float* __restrict__ X, const float* __restrict__ W,
    const float* __restrict__ bias, float* __restrict__ Y) {
  const int lane = threadIdx.x & 31;
  const int wave = threadIdx.x >> 5;
  const int m0   = blockIdx.x * 128 + wave * 16;
  const int n0   = blockIdx.y * 64;
  const int half = lane >> 4;   // selects K pair {0,1} vs {2,3}
  const int l15  = lane & 15;

  v8f acc0 = {}, acc1 = {}, acc2 = {}, acc3 = {};
  const float* arow = X + (size_t)(m0 + l15) * DM_;

  for (int k0 = 0; k0 < DM_; k0 += 4) {
    const int ka = k0 + half * 2;
    // A 16x4: lane<16 -> {A[l,k0],A[l,k0+1]}; lane>=16 -> K pair {k0+2,k0+3}
    v2f a = *(const v2f*)(arow + ka);
    // B 4x16: B[k,n] = W[n,k] -> contiguous float2 along k of row-major W
    v2f b0v = *(const v2f*)(W + (size_t)(n0 +  0 + l15) * DM_ + ka);
    v2f b1v = *(const v2f*)(W + (size_t)(n0 + 16 + l15) * DM_ + ka);
    v2f b2v = *(const v2f*)(W + (size_t)(n0 + 32 + l15) * DM_ + ka);
    v2f b3v = *(const v2f*)(W + (size_t)(n0 + 48 + l15) * DM_ + ka);
    acc0 = __builtin_amdgcn_wmma_f32_16x16x4_f32(false, a, false, b0v, (short)0, acc0, false, false);
    acc1 = __builtin_amdgcn_wmma_f32_16x16x4_f32(false, a, false, b1v, (short)0, acc1, false, false);
    acc2 = __builtin_amdgcn_wmma_f32_16x16x4_f32(false, a, false, b2v, (short)0, acc2, false, false);
    acc3 = __builtin_amdgcn_wmma_f32_16x16x4_f32(false, a, false, b3v, (short)0, acc3, false, false);
  }

  v8f accs[4] = {acc0, acc1, acc2, acc3};
#pragma unroll
  for (int j = 0; j < 4; ++j) {
    const int col = n0 + j * 16 + l15;
    const float bv = bias[col];
#pragma unroll
    for (int r = 0; r < 8; ++r) {
      // C/D 16x16: VGPR r holds M = 8*half + r, N = l15
      const int row = m0 + half * 8 + r;
      Y[(size_t)row * DM_ + col] = accs[j][r] + bv;
    }
  }
}

// ---------------------------------------------------------------------------
// RP[(bh*S + q)*21 + t] = dot(Q[b,q,h,:], rel_k_table[t,:])
// ---------------------------------------------------------------------------
__global__ void rp_kernel(const float* __restrict__ Q,
                          const float* __restrict__ relk,
                          float* __restrict__ RP) {
  int idx = blockIdx.x * blockDim.x + threadIdx.x;
  if (idx >= B_ * H_ * S_ * NT_) return;
  int t = idx % NT_;
  int r = idx / NT_;          // bh*S + q
  int q = r % S_;
  int bh = r / S_;
  int b = bh / H_, h = bh % H_;
  const float* qv = Q + ((size_t)(b * S_ + q)) * DM_ + h * DH_;
  const float* rk = relk + t * DH_;
  float s = 0.f;
#pragma unroll
  for (int d = 0; d < DH_; ++d) s += qv[d] * rk[d];
  RP[idx] = s;
}

// ---------------------------------------------------------------------------
// Partial K^T V:  MP[(bh*16+chunk)][i*32+j] = sum_{k in chunk} K[k,i]*V[k,j]
// ---------------------------------------------------------------------------
__global__ __launch_bounds__(256) void ktv_partial_kernel(
    const float* __restrict__ Kp, const float* __restrict__ Vp,
    float* __restrict__ MP) {
  const int bh = blockIdx.x, chunk = blockIdx.y;
  const int b = bh / H_, h = bh % H_;
  const int i  = threadIdx.x >> 3;        // 0..31
  const int j0 = (threadIdx.x & 7) * 4;   // 0..28
  const int CH = S_ / 16;                 // 128 keys per chunk
  float a0 = 0.f, a1 = 0.f, a2 = 0.f, a3 = 0.f;
  const int k0 = chunk * CH;
  for (int k = 0; k < CH; ++k) {
    const float* base = Kp + ((size_t)(b * S_ + k0 + k)) * DM_ + h * DH_;
    const float* vb   = Vp + ((size_t)(b * S_ + k0 + k)) * DM_ + h * DH_ + j0;
    float kv = base[i];
    a0 += kv * vb[0]; a1 += kv * vb[1]; a2 += kv * vb[2]; a3 += kv * vb[3];
  }
  float* m = MP + ((size_t)(bh * 16 + chunk)) * (DH_ * DH_) + i * DH_ + j0;
  m[0] = a0; m[1] = a1; m[2] = a2; m[3] = a3;
}

__global__ void ktv_reduce_kernel(const float* __restrict__ MP,
                                  float* __restrict__ M) {
  int idx = blockIdx.x * blockDim.x + threadIdx.x;
  if (idx >= B_ * H_ * DH_ * DH_) return;
  int bh = idx >> 10, e = idx & 1023;
  float s = 0.f;
#pragma unroll
  for (int c = 0; c < 16; ++c) s += MP[((size_t)(bh * 16 + c)) * 1024 + e];
  M[idx] = s;
}

// ---------------------------------------------------------------------------
// Row prefix sums per (b,h): P[bh][j][d] = sum_{k<j} src[b,k,h,d], j=0..S.
// 8 chunks of 256 rows, two-level scan.
// ---------------------------------------------------------------------------
__global__ __launch_bounds__(256) void prefix_kernel(
    const float* __restrict__ src, float* __restrict__ P) {
  const int bh = blockIdx.x;
  const int b = bh / H_, h = bh % H_;
  const int d = threadIdx.x & 31;
  const int chunk = threadIdx.x >> 5;  // 0..7
  const int CH = S_ / 8;               // 256
  __shared__ float csum[8][32];
  const int kstart = chunk * CH;
  float s = 0.f;
  for (int k = 0; k < CH; ++k)
    s += src[((size_t)(b * S_ + kstart + k)) * DM_ + h * DH_ + d];
  csum[chunk][d] = s;
  __syncthreads();
  float run = 0.f;
  for (int c = 0; c < chunk; ++c) run += csum[c][d];
  if (chunk == 0) P[((size_t)bh * (S_ + 1) + 0) * DH_ + d] = 0.f;
  for (int k = 0; k < CH; ++k) {
    run += src[((size_t)(b * S_ + kstart + k)) * DM_ + h * DH_ + d];
    P[((size_t)bh * (S_ + 1) + kstart + k + 1) * DH_ + d] = run;
  }
}

// ---------------------------------------------------------------------------
// Per (b,h,q): assemble weight1 + weight2 using the linear decomposition,
// write X[b,q,h*32+d] = (w1+w2)/16 (already in [b,s,h,d] transposed order).
// ---------------------------------------------------------------------------
__global__ __launch_bounds__(256) void combine_kernel(
    const float* __restrict__ Q, const float* __restrict__ Kp,
    const float* __restrict__ Vp, const float* __restrict__ M,
    const float* __restrict__ PK, const float* __restrict__ PV,
    const float* __restrict__ RP, const float* __restrict__ relv,
    float* __restrict__ X) {
  const int bh   = blockIdx.x >> 3;           // S/256 = 8 q-blocks per bh
  const int q    = (blockIdx.x & 7) * 256 + threadIdx.x;
  const int b = bh / H_, h = bh % H_;

  __shared__ float Ms[DH_ * DH_];   // 32x32 K^T V
  __shared__ float RVs[NT_ * DH_];  // rel_v_table
  for (int i = threadIdx.x; i < DH_ * DH_; i += 256) Ms[i] = M[(size_t)bh * 1024 + i];
  for (int i = threadIdx.x; i < NT_ * DH_; i += 256) RVs[i] = relv[i];
  __syncthreads();

  float qv[DH_], acc[DH_], rpl[NT_];
  const float* qptr = Q + ((size_t)(b * S_ + q)) * DM_ + h * DH_;
#pragma unroll
  for (int d = 0; d < DH_; ++d) qv[d] = qptr[d];
  const float* rp = RP + ((size_t)(bh * S_ + q)) * NT_;
#pragma unroll
  for (int t = 0; t < NT_; ++t) rpl[t] = rp[t];

  // term1: Q[q] @ (K^T V)
#pragma unroll
  for (int d = 0; d < DH_; ++d) acc[d] = 0.f;
  for (int i = 0; i < DH_; ++i) {
    float qi = qv[i];
#pragma unroll
    for (int d = 0; d < DH_; ++d) acc[d] += qi * Ms[i * DH_ + d];
  }

  // tail buckets via prefix sums of K
  int jlo = q - 9;  if (jlo < 0) jlo = 0;        // sum_{k<=q-10} = PK[max(q-9,0)]
  int jhi = q + 10; if (jhi > S_) jhi = S_;      // sum_{k>=q+10} = PK[S]-PK[jhi]
  float cnt0  = (q - 9  > 0) ? (float)(q - 9)        : 0.f;
  float cnt20 = (S_ - (q + 10) > 0) ? (float)(S_ - (q + 10)) : 0.f;
  const float* pk_lo = PK + ((size_t)bh * (S_ + 1) + jlo) * DH_;
  const float* pk_hi = PK + ((size_t)bh * (S_ + 1) + jhi) * DH_;
  const float* pk_S  = PK + ((size_t)bh * (S_ + 1) + S_) * DH_;
  float G0 = cnt0 * rpl[0], G20 = cnt20 * rpl[NT_ - 1];
#pragma unroll
  for (int d = 0; d < DH_; ++d) {
    G0  += qv[d] * pk_lo[d];
    G20 += qv[d] * (pk_S[d] - pk_hi[d]);
  }

  // middle buckets t=1..19 <-> dist = t-10 in [-9,9]: single band element each
  float G[19];
#pragma unroll
  for (int t = 0; t < 19; ++t) {
    int kk = q + t - 9;
    float g = 0.f;
    if (kk >= 0 && kk < S_) {
      const float* kr = Kp + ((size_t)(b * S_ + kk)) * DM_ + h * DH_;
#pragma unroll
      for (int d = 0; d < DH_; ++d) g += qv[d] * kr[d];
      g += rpl[t + 1];
    }
    G[t] = g;
  }

  // rel_v(weight1) tails + weight2 tails
  const float* pv_lo = PV + ((size_t)bh * (S_ + 1) + jlo) * DH_;
  const float* pv_hi = PV + ((size_t)bh * (S_ + 1) + jhi) * DH_;
  const float* pv_S  = PV + ((size_t)bh * (S_ + 1) + S_) * DH_;
#pragma unroll
  for (int d = 0; d < DH_; ++d) {
    float w = acc[d];
    w += rpl[0] * pv_lo[d] + rpl[NT_ - 1] * (pv_S[d] - pv_hi[d]);   // rel part of w1
    w += G0 * RVs[d] + G20 * RVs[(NT_ - 1) * DH_ + d];              // w2 tails
    acc[d] = w;
  }
  // band contributions: w1 rel middle (RP*V[kk]) + w2 middle (G_t*rel_v[t])
#pragma unroll
  for (int t = 0; t < 19; ++t) {
    int kk = q + t - 9;
    if (kk >= 0 && kk < S_) {
      const float* vr = Vp + ((size_t)(b * S_ + kk)) * DM_ + h * DH_;
      float rt = rpl[t + 1], gt = G[t];
#pragma unroll
      for (int d = 0; d < DH_; ++d)
        acc[d] += rt * vr[d] + gt * RVs[(t + 1) * DH_ + d];
    }
  }

  const float inv = 1.0f / 16.0f;  // SCALE = sqrt(256)
  float* xp = X + ((size_t)(b * S_ + q)) * DM_ + h * DH_;
#pragma unroll
  for (int d = 0; d < DH_; ++d) xp[d] = acc[d] * inv;
}

// ---------------------------------------------------------------------------
extern "C" void kernel_launch(void* const* d_in, const int* in_sizes, int n_in,
                              void* d_out, int out_size, void* d_ws, size_t ws_size,
                              hipStream_t stream) {
  (void)in_sizes; (void)n_in; (void)out_size; (void)ws_size;
  const float* q    = (const float*)d_in[0];
  const float* k    = (const float*)d_in[1];
  const float* v    = (const float*)d_in[2];
  /* d_in[3] = mask (no-op) */
  const float* Wq   = (const float*)d_in[4];
  const float* bq   = (const float*)d_in[5];
  const float* Wk   = (const float*)d_in[6];
  const float* bk   = (const float*)d_in[7];
  const float* Wv   = (const float*)d_in[8];
  const float* bv   = (const float*)d_in[9];
  const float* W0   = (const float*)d_in[10];
  const float* b0   = (const float*)d_in[11];
  const float* relk = (const float*)d_in[12];
  const float* relv = (const float*)d_in[13];
  float* out = (float*)d_out;
  float* ws  = (float*)d_ws;

  const size_t NQ = (size_t)B_ * S_ * DM_;            // 1,048,576
  float* Q  = ws;
  float* Kp = Q  + NQ;
  float* Vp = Kp + NQ;
  float* X  = Vp + NQ;
  float* RP = X  + NQ;                                // B*H*S*21
  float* MP = RP + (size_t)B_ * H_ * S_ * NT_;        // 16*16*1024
  float* M  = MP + (size_t)B_ * H_ * 16 * 1024;       // 16*1024
  float* PK = M  + (size_t)B_ * H_ * 1024;            // 16*(S+1)*32
  float* PV = PK + (size_t)B_ * H_ * (S_ + 1) * DH_;

  dim3 pg(B_ * S_ / 128, DM_ / 64);   // (32, 4)

  proj_wmma_kernel<<<pg, 256, 0, stream>>>(q, Wq, bq, Q);
  proj_wmma_kernel<<<pg, 256, 0, stream>>>(k, Wk, bk, Kp);
  proj_wmma_kernel<<<pg, 256, 0, stream>>>(v, Wv, bv, Vp);

  rp_kernel<<<(B_ * H_ * S_ * NT_ + 255) / 256, 256, 0, stream>>>(Q, relk, RP);

  ktv_partial_kernel<<<dim3(B_ * H_, 16), 256, 0, stream>>>(Kp, Vp, MP);
  ktv_reduce_kernel<<<(B_ * H_ * 1024 + 255) / 256, 256, 0, stream>>>(MP, M);

  prefix_kernel<<<B_ * H_, 256, 0, stream>>>(Kp, PK);
  prefix_kernel<<<B_ * H_, 256, 0, stream>>>(Vp, PV);

  combine_kernel<<<B_ * H_ * (S_ / 256), 256, 0, stream>>>(
      Q, Kp, Vp, M, PK, PV, RP, relv, X);

  proj_wmma_kernel<<<pg, 256, 0, stream>>>(X, W0, b0, out);
}